// MONPeacemanRachford_7705171329223
// MI455X (gfx1250) — compile-verified
//
#include <hip/hip_runtime.h>

typedef __attribute__((ext_vector_type(2))) float v2f;
typedef __attribute__((ext_vector_type(8))) float v8f;

#define TILE_LD 132                   // padded A-staging row stride (floats)
#define SV_FLOATS (8 * 16 * TILE_LD)  // 8 waves x 16 rows x 132
#define PRLD 144                      // B pair-row stride: 2*144 % 64 == 32 banks
#define BB_FLOATS (64 * PRLD * 2)     // 64 K-pairs x 144 cols x 2 floats
#define PREP_W 257                    // augmented matrix padded stride
#define MARGIN 0.05f
#define N_ITER 50

// ---------------------------------------------------------------------------
// Prep: wT[k*128+n] = W[n][k];  winvT[k*128+n] = inv(2I - W)[n][k]
// One block of 1024 threads, Gauss-Jordan in LDS (well conditioned, no pivot).
// ---------------------------------------------------------------------------
__global__ __launch_bounds__(1024)
void mon_prep_kernel(const float* __restrict__ A, const float* __restrict__ B,
                     float* __restrict__ winvT, float* __restrict__ wT) {
  extern __shared__ float smem[];
  float* As   = smem;              // 128*128
  float* Maug = smem + 128 * 128;  // 128*257
  const int tid = threadIdx.x;

  for (int i = tid; i < 128 * 128; i += 1024) As[i] = A[i];
  __syncthreads();

  // Phase A: W[n][k] = (1-m)*d(n,k) - sum_t A[t][n]A[t][k] + B[n][k] - B[k][n]
  for (int q = 0; q < 16; ++q) {
    int idx = tid + q * 1024;
    int k = idx >> 7, n = idx & 127;
    float acc = 0.0f;
    for (int t = 0; t < 128; ++t)
      acc = fmaf(As[t * 128 + n], As[t * 128 + k], acc);
    float w = -acc + B[n * 128 + k] - B[k * 128 + n];
    if (n == k) w += (1.0f - MARGIN);
    wT[idx] = w;  // wT[k*128+n] = W[n][k]
  }
  __syncthreads();

  // Phase B: Maug = [2I - W | I], Gauss-Jordan
  const int row  = tid >> 3;  // 0..127
  const int colg = tid & 7;   // 8 col-groups of 32
  for (int jj = 0; jj < 32; ++jj) {
    int j = colg * 32 + jj;
    float v;
    if (j < 128)
      v = ((row == j) ? 2.0f : 0.0f) - wT[j * 128 + row];  // W[row][j]
    else
      v = ((j - 128) == row) ? 1.0f : 0.0f;
    Maug[row * PREP_W + j] = v;
  }

  for (int k = 0; k < 128; ++k) {
    __syncthreads();
    float pinv = 1.0f / Maug[k * PREP_W + k];
    __syncthreads();
    if (row == k)
      for (int jj = 0; jj < 32; ++jj) {
        int j = colg * 32 + jj;
        Maug[k * PREP_W + j] *= pinv;
      }
    __syncthreads();
    float f = Maug[row * PREP_W + k];
    __syncthreads();
    if (row != k)
      for (int jj = 0; jj < 32; ++jj) {
        int j = colg * 32 + jj;
        Maug[row * PREP_W + j] =
            fmaf(-f, Maug[k * PREP_W + j], Maug[row * PREP_W + j]);
      }
  }
  __syncthreads();

  // winvT[k*128+n] = Winv[n][k] = Maug[n][128+k]
  for (int q = 0; q < 16; ++q) {
    int idx = tid + q * 1024;
    int k = idx >> 7, n = idx & 127;
    winvT[idx] = Maug[n * PREP_W + 128 + k];
  }
}

// ---------------------------------------------------------------------------
// Load a 128x128 B-matrix (already stored transposed in global: g[k*128+n])
// into LDS in K-pair-interleaved layout:
//   BbP[((k>>1)*PRLD + n)*2 + (k&1)] = g[k*128+n]
// so each WMMA B fragment is one contiguous ds_load_b64.
// ---------------------------------------------------------------------------
__device__ __forceinline__ void load_bmat(float* __restrict__ BbP,
                                          const float* __restrict__ g,
                                          int tid) {
  for (int q = 0; q < 64; ++q) {
    int i = tid + q * 256;
    int k = i >> 7, n = i & 127;
    BbP[((k >> 1) * PRLD + n) * 2 + (k & 1)] = g[i];
  }
}

// ---------------------------------------------------------------------------
// Tile GEMM: D(16x128) += Sv-tile(16x128) @ B(128x128), 32 K-steps of
// v_wmma_f32_16x16x4_f32 over 8 column blocks.
// A layout (ISA): lanes 0-15 hold K{4k,4k+1}, lanes 16-31 hold K{4k+2,4k+3}
// of row lane%16. B fragment: pair rows {row0,row0+1} at col n*16+lane%16.
// ---------------------------------------------------------------------------
__device__ __forceinline__ void tile_gemm(const float* __restrict__ Sv,
                                          const float* __restrict__ BbP,
                                          int svBase, int lane, v8f acc[8]) {
  const int l16 = lane & 15;
  const int hi  = lane >> 4;           // 0: K{0,1}, 1: K{2,3} of each step
  const float* ab = Sv + svBase + l16 * TILE_LD + 2 * hi;
  const float* bb = BbP + hi * (PRLD * 2) + 2 * l16;  // k2 = 2k + hi
#pragma unroll 4
  for (int k = 0; k < 32; ++k) {
    v2f a = *(const v2f*)(ab + 4 * k);
    const float* bk = bb + k * (2 * PRLD * 2);
#pragma unroll
    for (int n = 0; n < 8; ++n) {
      v2f b = *(const v2f*)(bk + n * 32);
      acc[n] = __builtin_amdgcn_wmma_f32_16x16x4_f32(
          false, a, false, b, (short)0, acc[n], false, false);
    }
  }
}

// ---------------------------------------------------------------------------
// Main persistent-tile kernel: 8 waves/block, each wave owns 16 rows.
// bias & u live in VGPRs across all 50 PR iterations.
// ---------------------------------------------------------------------------
__global__ __launch_bounds__(256)
void mon_pr_kernel(const float* __restrict__ x, const float* __restrict__ Uw,
                   const float* __restrict__ ub,
                   const float* __restrict__ winvT,
                   const float* __restrict__ wT, float* __restrict__ out) {
  extern __shared__ float smem[];
  float* Sv  = smem;              // per-wave staging tiles
  float* BbP = smem + SV_FLOATS;  // shared pair-interleaved B-matrix

  const int tid   = threadIdx.x;
  const int lane  = tid & 31;
  const int wave  = tid >> 5;
  const int l16   = lane & 15;
  const int rowSt = (lane >> 4) * 8;  // C/D layout: VGPR r -> rows r / r+8
  const int svBase = wave * 16 * TILE_LD;
  const long tileRow0 = (long)blockIdx.x * 128 + wave * 16;

  // ---- B := Uw^T  (element (k,n) = Uw[n][k]) ----
  for (int q = 0; q < 64; ++q) {
    int i = tid + q * 256;
    int k = i >> 7, n = i & 127;
    BbP[((k >> 1) * PRLD + n) * 2 + (k & 1)] = Uw[n * 128 + k];
  }
  // ---- stage x tile (per-wave private region) ----
  {
    const int r  = lane >> 1;
    const int c0 = (lane & 1) * 64;
    const float* xr = x + (tileRow0 + r) * 128 + c0;
    float* sr = Sv + svBase + r * TILE_LD + c0;
#pragma unroll
    for (int q = 0; q < 16; ++q)
      *(float4*)(sr + q * 4) = *(const float4*)(xr + q * 4);
  }
  __syncthreads();

  // ---- bias = x @ Uw^T + ub ----
  float bias[8][8];
  {
    v8f acc[8];
#pragma unroll
    for (int n = 0; n < 8; ++n)
#pragma unroll
      for (int r = 0; r < 8; ++r) acc[n][r] = 0.0f;
    tile_gemm(Sv, BbP, svBase, lane, acc);
#pragma unroll
    for (int f = 0; f < 8; ++f) {
      float ubv = ub[f * 16 + l16];
#pragma unroll
      for (int r = 0; r < 8; ++r) bias[f][r] = acc[f][r] + ubv;
    }
  }

  // ---- B := Winv^T ----
  __syncthreads();
  load_bmat(BbP, winvT, tid);
  __syncthreads();

  // ---- Peaceman-Rachford: z = relu(u); u12 = 2z-u; z12 = (u12+bias)@Winv^T;
  //      u = 2*z12 - u12 ----
  float u[8][8];
#pragma unroll
  for (int f = 0; f < 8; ++f)
#pragma unroll
    for (int r = 0; r < 8; ++r) u[f][r] = 0.0f;

  for (int it = 0; it < N_ITER; ++it) {
    // stage v = u12 + bias; u := u12   (LDS ops per-wave are in-order)
#pragma unroll
    for (int f = 0; f < 8; ++f)
#pragma unroll
      for (int r = 0; r < 8; ++r) {
        float uu  = u[f][r];
        float u12 = 2.0f * fmaxf(uu, 0.0f) - uu;
        u[f][r] = u12;
        Sv[svBase + (r + rowSt) * TILE_LD + f * 16 + l16] = u12 + bias[f][r];
      }
    v8f acc[8];
#pragma unroll
    for (int n = 0; n < 8; ++n)
#pragma unroll
      for (int r = 0; r < 8; ++r) acc[n][r] = 0.0f;
    tile_gemm(Sv, BbP, svBase, lane, acc);
#pragma unroll
    for (int f = 0; f < 8; ++f)
#pragma unroll
      for (int r = 0; r < 8; ++r) u[f][r] = 2.0f * acc[f][r] - u[f][r];
  }

  // ---- B := W^T ----
  __syncthreads();
  load_bmat(BbP, wT, tid);
  __syncthreads();

  // ---- out = relu(bias + relu(u) @ W^T) ----
#pragma unroll
  for (int f = 0; f < 8; ++f)
#pragma unroll
    for (int r = 0; r < 8; ++r)
      Sv[svBase + (r + rowSt) * TILE_LD + f * 16 + l16] = fmaxf(u[f][r], 0.0f);
  v8f acc[8];
#pragma unroll
  for (int n = 0; n < 8; ++n)
#pragma unroll
    for (int r = 0; r < 8; ++r) acc[n][r] = 0.0f;
  tile_gemm(Sv, BbP, svBase, lane, acc);
#pragma unroll
  for (int f = 0; f < 8; ++f)
#pragma unroll
    for (int r = 0; r < 8; ++r) {
      float o = fmaxf(bias[f][r] + acc[f][r], 0.0f);
      out[(tileRow0 + r + rowSt) * 128 + f * 16 + l16] = o;
    }
}

// ---------------------------------------------------------------------------
extern "C" void kernel_launch(void* const* d_in, const int* in_sizes, int n_in,
                              void* d_out, int out_size, void* d_ws,
                              size_t ws_size, hipStream_t stream) {
  const float* x  = (const float*)d_in[0];
  const float* Uw = (const float*)d_in[1];
  const float* ub = (const float*)d_in[2];
  const float* A  = (const float*)d_in[3];
  const float* B  = (const float*)d_in[4];
  float* out = (float*)d_out;

  float* ws    = (float*)d_ws;
  float* winvT = ws;          // 16384 floats
  float* wT    = ws + 16384;  // 16384 floats

  const size_t prepLds = (size_t)(128 * 128 + 128 * PREP_W) * sizeof(float);
  const size_t mainLds = (size_t)(SV_FLOATS + BB_FLOATS) * sizeof(float);

  (void)hipFuncSetAttribute((const void*)mon_prep_kernel,
                            hipFuncAttributeMaxDynamicSharedMemorySize,
                            (int)prepLds);
  (void)hipFuncSetAttribute((const void*)mon_pr_kernel,
                            hipFuncAttributeMaxDynamicSharedMemorySize,
                            (int)mainLds);

  mon_prep_kernel<<<1, 1024, prepLds, stream>>>(A, B, winvT, wT);

  const int rows = in_sizes[0] / 128;  // 65536
  mon_pr_kernel<<<rows / 128, 256, mainLds, stream>>>(x, Uw, ub, winvT, wT,
                                                      out);
}